// Postprocessor_39616778338421
// MI455X (gfx1250) — compile-verified
//
#include <hip/hip_runtime.h>
#include <stdint.h>

// Problem constants (match reference)
#define BATCH      32
#define NANCH      8400
#define NCLS       80
#define ROWLEN     85          // 5 + NUM_CLASSES floats per anchor
#define TOPKN      1000
#define NSORT      16384       // next pow2 >= NANCH for bitonic sort
#define CONF_THRE  0.25f
#define NMS_THRE   0.45f
#define CLS_OFF    10.0f

#define ANCH_PER_BLK 128
#define DEC_THREADS  256

// ---------------------------------------------------------------------------
// CDNA5 async data mover: global -> LDS, tracked by ASYNCcnt.
// GVS form: addr = SGPR64 base + VGPR32 byte offset. VDST vgpr holds the
// per-lane LDS byte address (low 32 bits of the flat shared pointer).
// ---------------------------------------------------------------------------
__device__ __forceinline__ void async_load_b128_to_lds(unsigned lds_byte_addr,
                                                       unsigned gbyte_off,
                                                       const float* sbase) {
  asm volatile("global_load_async_to_lds_b128 %0, %1, %2"
               :: "v"(lds_byte_addr), "v"(gbyte_off), "s"(sbase)
               : "memory");
}
__device__ __forceinline__ void wait_asynccnt0() {
  asm volatile("s_wait_asynccnt 0" ::: "memory");
}

// ---------------------------------------------------------------------------
// Kernel 1: decode.  Each block stages 128 contiguous anchor rows
// (128*85*4 = 43520 B) into LDS with async b128 copies (fully coalesced),
// then one thread per anchor does box decode + class max/argmax from LDS.
// LDS reads: addr = 85*t + k dwords -> bank (21*t + k) mod 64, gcd(21,64)=1
// -> conflict free across a wave.
// ---------------------------------------------------------------------------
__global__ __launch_bounds__(DEC_THREADS)
void decode_kernel(const float* __restrict__ pred,
                   float* __restrict__ scores,   // masked scores (invalid -> -1)
                   float* __restrict__ boxes,    // [B*N][4] x1,y1,x2,y2
                   int*   __restrict__ cls) {
  __shared__ float smem[ANCH_PER_BLK * ROWLEN];   // 43520 B
  const int tid = threadIdx.x;
  const int blk = blockIdx.x;
  const float* gbase = pred + (size_t)blk * (ANCH_PER_BLK * ROWLEN);

  const int n128 = (ANCH_PER_BLK * ROWLEN) / 4;   // 2720 b128 packets
  for (int u = tid; u < n128; u += DEC_THREADS) {
    unsigned lds = (unsigned)(unsigned long long)(const void*)&smem[u * 4];
    async_load_b128_to_lds(lds, (unsigned)(u * 16), gbase);
  }
  wait_asynccnt0();
  __syncthreads();

  if (tid < ANCH_PER_BLK) {
    const float* row = &smem[tid * ROWLEN];
    const float cx = row[0], cy = row[1];
    const float w2 = row[2] * 0.5f, h2 = row[3] * 0.5f;
    const float obj = row[4];
    float best = row[5];
    int   arg  = 0;
#pragma unroll 4
    for (int c = 1; c < NCLS; ++c) {
      const float v = row[5 + c];
      if (v > best) { best = v; arg = c; }   // strict > == first-max (argmax)
    }
    const float sc     = obj * best;
    const float masked = (sc >= CONF_THRE) ? sc : -1.0f;
    const size_t g = (size_t)blk * ANCH_PER_BLK + tid;
    scores[g]      = masked;
    boxes[g*4 + 0] = cx - w2;
    boxes[g*4 + 1] = cy - h2;
    boxes[g*4 + 2] = cx + w2;
    boxes[g*4 + 3] = cy + h2;
    cls[g]         = arg;
  }
}

// ---------------------------------------------------------------------------
// Kernel 2: per-batch top-k via full in-LDS bitonic sort of 16384 64-bit keys
// (128 KB dynamic LDS -- legal on a 320 KB/WGP CDNA5 workgroup).
// key = (orderedFloat(maskedScore) << 32) | ~index  => descending sort gives
// exactly lax.top_k order incl. stable index-ascending tie break.
// ---------------------------------------------------------------------------
__global__ __launch_bounds__(1024)
void topk_kernel(const float* __restrict__ scores,
                 int* __restrict__ topidx) {
  extern __shared__ unsigned long long keys[];   // NSORT * 8 = 131072 B
  const int b   = blockIdx.x;
  const int tid = threadIdx.x;

  for (int i = tid; i < NSORT; i += 1024) {
    unsigned long long k = 0ull;                 // padding sorts last
    if (i < NANCH) {
      const float s = scores[(size_t)b * NANCH + i];
      unsigned u = __float_as_uint(s);
      u = (u & 0x80000000u) ? ~u : (u | 0x80000000u);  // order-preserving map
      k = ((unsigned long long)u << 32) | (unsigned)(~(unsigned)i);
    }
    keys[i] = k;
  }

  for (unsigned size = 2; size <= NSORT; size <<= 1) {
    for (unsigned stride = size >> 1; stride > 0; stride >>= 1) {
      __syncthreads();
      for (unsigned t = tid; t < NSORT / 2; t += 1024) {
        const unsigned pos = 2u * t - (t & (stride - 1u));
        const bool ddd = ((t & (size >> 1)) != 0u);   // dir=0 => descending
        const unsigned long long a = keys[pos];
        const unsigned long long c = keys[pos + stride];
        if ((a > c) == ddd) { keys[pos] = c; keys[pos + stride] = a; }
      }
    }
  }
  __syncthreads();

  for (int i = tid; i < TOPKN; i += 1024)
    topidx[(size_t)b * TOPKN + i] = (int)(~(unsigned)(keys[i] & 0xFFFFFFFFull));
}

// ---------------------------------------------------------------------------
// Kernel 3: per-batch greedy NMS + output. One 1024-thread block per batch.
// Class-offset boxes + areas + keep flags live in LDS; the sequential greedy
// loop broadcasts keep[i] (uniform LDS read) and 1000 lanes clear keep[j>i].
// ---------------------------------------------------------------------------
__global__ __launch_bounds__(1024)
void nms_kernel(const float* __restrict__ scores,
                const float* __restrict__ boxes,
                const int*   __restrict__ cls,
                const int*   __restrict__ topidx,
                float* __restrict__ out) {       // dets (B*K*6) then keep (B*K)
  __shared__ float X1[TOPKN], Y1[TOPKN], X2[TOPKN], Y2[TOPKN], AREA[TOPKN];
  __shared__ float RB0[TOPKN], RB1[TOPKN], RB2[TOPKN], RB3[TOPKN], SC[TOPKN];
  __shared__ int   CL[TOPKN], KEEP[TOPKN];

  const int b   = blockIdx.x;
  const int tid = threadIdx.x;

  if (tid < TOPKN) {
    const int idx = topidx[(size_t)b * TOPKN + tid];
    const size_t g = (size_t)b * NANCH + idx;
    const float s  = scores[g];
    const int   c  = cls[g];
    const float b0 = boxes[g*4+0], b1 = boxes[g*4+1];
    const float b2 = boxes[g*4+2], b3 = boxes[g*4+3];
    const float off = (float)c * CLS_OFF;
    RB0[tid] = b0; RB1[tid] = b1; RB2[tid] = b2; RB3[tid] = b3;
    X1[tid] = b0 + off; Y1[tid] = b1 + off;
    X2[tid] = b2 + off; Y2[tid] = b3 + off;
    AREA[tid] = (b2 - b0) * (b3 - b1);
    SC[tid]  = s;
    CL[tid]  = c;
    KEEP[tid] = (s >= CONF_THRE) ? 1 : 0;
  }
  __syncthreads();

  for (int i = 0; i < TOPKN; ++i) {
    const int ki = KEEP[i];                      // uniform across block
    if (ki) {
      const int j = tid;
      if (j < TOPKN && j > i && KEEP[j]) {
        const float ix1 = fmaxf(X1[i], X1[j]);
        const float iy1 = fmaxf(Y1[i], Y1[j]);
        const float ix2 = fminf(X2[i], X2[j]);
        const float iy2 = fminf(Y2[i], Y2[j]);
        const float inter = fmaxf(ix2 - ix1, 0.0f) * fmaxf(iy2 - iy1, 0.0f);
        const float uni   = AREA[i] + AREA[j] - inter;
        const float iou   = inter / fmaxf(uni, 1e-9f);
        if (iou > NMS_THRE) KEEP[j] = 0;
      }
    }
    __syncthreads();
  }

  if (tid < TOPKN) {
    const bool kp = (KEEP[tid] != 0);
    const size_t base = ((size_t)b * TOPKN + tid) * 6;
    out[base + 0] = kp ? RB0[tid] : 0.0f;
    out[base + 1] = kp ? RB1[tid] : 0.0f;
    out[base + 2] = kp ? RB2[tid] : 0.0f;
    out[base + 3] = kp ? RB3[tid] : 0.0f;
    out[base + 4] = kp ? SC[tid]  : 0.0f;
    out[base + 5] = kp ? (float)CL[tid] : 0.0f;
    out[(size_t)BATCH * TOPKN * 6 + (size_t)b * TOPKN + tid] = kp ? 1.0f : 0.0f;
  }
}

// ---------------------------------------------------------------------------
extern "C" void kernel_launch(void* const* d_in, const int* in_sizes, int n_in,
                              void* d_out, int out_size, void* d_ws, size_t ws_size,
                              hipStream_t stream) {
  (void)in_sizes; (void)n_in; (void)out_size; (void)ws_size;

  const float* pred = (const float*)d_in[0];
  float* out = (float*)d_out;

  // Workspace carve-up (~6.6 MB)
  float* ws_scores = (float*)d_ws;                       // BATCH*NANCH
  float* ws_boxes  = ws_scores + (size_t)BATCH * NANCH;  // BATCH*NANCH*4
  int*   ws_cls    = (int*)(ws_boxes + (size_t)BATCH * NANCH * 4);
  int*   ws_topidx = ws_cls + (size_t)BATCH * NANCH;     // BATCH*TOPKN

  const int nblk = (BATCH * NANCH) / ANCH_PER_BLK;       // 2100 exactly
  decode_kernel<<<nblk, DEC_THREADS, 0, stream>>>(pred, ws_scores, ws_boxes, ws_cls);
  topk_kernel<<<BATCH, 1024, NSORT * sizeof(unsigned long long), stream>>>(ws_scores, ws_topidx);
  nms_kernel<<<BATCH, 1024, 0, stream>>>(ws_scores, ws_boxes, ws_cls, ws_topidx, out);
}